// Rec_SetCriterion_41901700940189
// MI455X (gfx1250) — compile-verified
//
#include <hip/hip_runtime.h>
#include <hip/hip_bf16.h>
#include <math.h>

// ---------------------------------------------------------------------------
// Shapes (fixed by the harness / reference setup_inputs)
// ---------------------------------------------------------------------------
#define BS    128      // batch
#define CREC  107      // rec classes
#define CLOC  25       // loc classes
#define QN    1024     // queries = 32*32
#define NT    25       // targets per batch
#define TPB   256      // hungarian block (8 wave32)
#define CE_TPB 128     // CE block

#define F_ALPHA 0.25f
#define F_EPS   1e-8f
#define BIGINF  1e18

// ---------------------------------------------------------------------------
// CDNA5 async global->LDS path (builtin confirmed present; param 0 is v4i*)
// ---------------------------------------------------------------------------
#if defined(__has_builtin)
#  if __has_builtin(__builtin_amdgcn_global_load_async_to_lds_b128)
#    define HAVE_ASYNC_B128 1
#  endif
#  if __has_builtin(__builtin_amdgcn_s_wait_asynccnt)
#    define HAVE_WAIT_ASYNC 1
#  endif
#endif

typedef int v4i __attribute__((vector_size(4 * sizeof(int))));
typedef __attribute__((address_space(1))) v4i as1_v4i;
typedef __attribute__((address_space(3))) v4i as3_v4i;

__device__ __forceinline__ as1_v4i* glob_cast_v4(const void* p) {
  return (as1_v4i*)(unsigned long long)(p);
}
__device__ __forceinline__ as3_v4i* lds_cast_v4(void* p) {
  // generic LDS pointer: low 32 bits are the LDS byte offset (ISA 10.2)
  return (as3_v4i*)(unsigned int)(unsigned long long)(p);
}

__device__ __forceinline__ void wait_async0() {
#if defined(HAVE_WAIT_ASYNC)
  __builtin_amdgcn_s_wait_asynccnt(0);
#else
  asm volatile("s_wait_asynccnt 0" ::: "memory");
#endif
}

// ---------------------------------------------------------------------------
// focal matcher cost: pos - neg for p = sigmoid(x), gamma = 2
// ---------------------------------------------------------------------------
__device__ __forceinline__ float focal_cost(float x) {
  float p  = 1.0f / (1.0f + expf(-x));
  float om = 1.0f - p;
  float neg = (1.0f - F_ALPHA) * p * p * (-logf(om + F_EPS));
  float pos = F_ALPHA * om * om * (-logf(p + F_EPS));
  return pos - neg;
}

// ---------------------------------------------------------------------------
// Kernel 1: per-batch cost build (in LDS) + Jonker-Volgenant assignment.
// One workgroup per batch. cost kept LDS-resident (re-read ~25x by Dijkstra).
// Mirrors the reference _lsa exactly (transposed: n=NT rows, m=QN cols, f64).
// Output: match[b*NT + t] = query index assigned to target t.
// ---------------------------------------------------------------------------
__global__ void __launch_bounds__(TPB)
hungarian_kernel(const float* __restrict__ rec,
                 const float* __restrict__ loc,
                 const int*   __restrict__ targets,
                 int*         __restrict__ match)
{
  __shared__ __align__(16) float cost[NT * QN];   // 100 KB
  __shared__ double v_s[QN + 1];                  // 8.2 KB
  __shared__ double minv_s[QN + 1];               // 8.2 KB
  __shared__ double u_s[NT + 1];
  __shared__ int    p_s[QN + 1];
  __shared__ int    way_s[QN + 1];
  __shared__ unsigned char used_s[QN + 1];
  __shared__ double red_v[TPB];
  __shared__ int    red_j[TPB];
  __shared__ double bc_delta;
  __shared__ int    bc_j1;

  const int b   = blockIdx.x;
  const int tid = threadIdx.x;

  // ---- stage gathered rec rows (class = targets[b][j]) into LDS ----------
  // one 16B async transfer per thread per row: 256 * 16B = 4 KB = full row
  for (int j = 0; j < NT; ++j) {
    const int cls = targets[b * NT + j];
    const float* src = rec + ((size_t)b * CREC + (size_t)cls) * QN + tid * 4;
    float* dst = &cost[j * QN + tid * 4];
#if defined(HAVE_ASYNC_B128)
    __builtin_amdgcn_global_load_async_to_lds_b128(glob_cast_v4(src),
                                                   lds_cast_v4(dst),
                                                   /*offset=*/0, /*cpol=*/0);
#else
    *(float4*)dst = *(const float4*)src;
#endif
  }
#if defined(HAVE_ASYNC_B128)
  wait_async0();
#endif
  __syncthreads();

  // ---- cost[j][q] = 2*focal(rec[b, t_j, q]) + focal(loc[b, j, q]) ---------
  for (int idx = tid; idx < NT * QN; idx += TPB) {
    const int j = idx >> 10;           // / QN
    const int q = idx & (QN - 1);      // % QN
    const float cr = cost[idx];
    const float cl = loc[((size_t)b * CLOC + j) * QN + q];
    cost[idx] = 2.0f * focal_cost(cr) + focal_cost(cl);
  }

  // ---- JV init ------------------------------------------------------------
  for (int j = tid; j <= QN; j += TPB) { v_s[j] = 0.0; p_s[j] = 0; way_s[j] = 0; }
  if (tid <= NT) u_s[tid] = 0.0;
  __syncthreads();

  for (int i = 1; i <= NT; ++i) {
    if (tid == 0) p_s[0] = i;
    for (int j = tid; j <= QN; j += TPB) { minv_s[j] = BIGINF; used_s[j] = 0; }
    __syncthreads();

    int j0 = 0;
    while (true) {
      if (tid == 0) used_s[j0] = 1;
      __syncthreads();

      const int    i0   = p_s[j0];
      const double ui0  = u_s[i0];
      const float* crow = &cost[(i0 - 1) * QN];

      // relax free columns
      for (int j = 1 + tid; j <= QN; j += TPB) {
        if (!used_s[j]) {
          const double cur = (double)crow[j - 1] - ui0 - v_s[j];
          if (cur < minv_s[j]) { minv_s[j] = cur; way_s[j] = j0; }
        }
      }
      __syncthreads();

      // argmin over free columns; (value, index) order == np.argmin first-min
      double bv = BIGINF; int bj = QN + 1;
      for (int j = 1 + tid; j <= QN; j += TPB) {
        if (!used_s[j]) {
          const double x = minv_s[j];
          if (x < bv || (x == bv && j < bj)) { bv = x; bj = j; }
        }
      }
      red_v[tid] = bv; red_j[tid] = bj;
      __syncthreads();
      for (int s = TPB / 2; s > 0; s >>= 1) {
        if (tid < s) {
          const double ov = red_v[tid + s]; const int oj = red_j[tid + s];
          if (ov < red_v[tid] || (ov == red_v[tid] && oj < red_j[tid])) {
            red_v[tid] = ov; red_j[tid] = oj;
          }
        }
        __syncthreads();
      }
      if (tid == 0) { bc_delta = red_v[0]; bc_j1 = red_j[0]; }
      __syncthreads();
      const double delta = bc_delta;
      const int    j1    = bc_j1;

      // dual updates (u indices p_s[j] are distinct over used j -> no race)
      for (int j = tid; j <= QN; j += TPB) {
        if (used_s[j]) { u_s[p_s[j]] += delta; v_s[j] -= delta; }
        else if (j >= 1) { minv_s[j] -= delta; }
      }
      __syncthreads();

      j0 = j1;
      if (p_s[j0] == 0) break;
    }

    // augment along way[] (serial, tiny)
    if (tid == 0) {
      int j = j0;
      while (j != 0) { const int jp = way_s[j]; p_s[j] = p_s[jp]; j = jp; }
    }
    __syncthreads();
  }

  // invert: row r (1..NT) assigned to column j -> query j-1 matched to tgt r-1
  for (int j = 1 + tid; j <= QN; j += TPB) {
    const int r = p_s[j];
    if (r > 0) match[b * NT + (r - 1)] = j - 1;
  }
}

// ---------------------------------------------------------------------------
// Kernel 2: per matched row cross-entropy (rec: 107 classes, loc: 25 classes)
// one block per (b, t) row; logsumexp via LDS tree reductions.
// ---------------------------------------------------------------------------
__global__ void __launch_bounds__(CE_TPB)
ce_kernel(const float* __restrict__ rec, const float* __restrict__ loc,
          const int* __restrict__ targets, const int* __restrict__ match,
          float* __restrict__ rloss, float* __restrict__ ploss)
{
  __shared__ float red[CE_TPB];
  const int r   = blockIdx.x;            // 0..BS*NT-1
  const int b   = r / NT;
  const int t   = r - b * NT;
  const int tid = threadIdx.x;
  const int q   = match[r];
  const int lab = targets[r];

  // ---- rec CE -------------------------------------------------------------
  const float* rb = rec + (size_t)b * CREC * QN + q;
  __builtin_prefetch(rb, 0, 1);          // global_prefetch_b8 on gfx1250
  const float x = (tid < CREC) ? rb[(size_t)tid * QN] : -3.4e38f;
  red[tid] = x; __syncthreads();
  for (int s = CE_TPB / 2; s > 0; s >>= 1) {
    if (tid < s) red[tid] = fmaxf(red[tid], red[tid + s]);
    __syncthreads();
  }
  const float m = red[0]; __syncthreads();
  red[tid] = (tid < CREC) ? expf(x - m) : 0.0f; __syncthreads();
  for (int s = CE_TPB / 2; s > 0; s >>= 1) {
    if (tid < s) red[tid] += red[tid + s];
    __syncthreads();
  }
  if (tid == 0) rloss[r] = logf(red[0]) + m - rb[(size_t)lab * QN];
  __syncthreads();

  // ---- loc CE (label = t) -------------------------------------------------
  const float* lb = loc + (size_t)b * CLOC * QN + q;
  const float y = (tid < CLOC) ? lb[(size_t)tid * QN] : -3.4e38f;
  red[tid] = y; __syncthreads();
  for (int s = CE_TPB / 2; s > 0; s >>= 1) {
    if (tid < s) red[tid] = fmaxf(red[tid], red[tid + s]);
    __syncthreads();
  }
  const float m2 = red[0]; __syncthreads();
  red[tid] = (tid < CLOC) ? expf(y - m2) : 0.0f; __syncthreads();
  for (int s = CE_TPB / 2; s > 0; s >>= 1) {
    if (tid < s) red[tid] += red[tid + s];
    __syncthreads();
  }
  if (tid == 0) ploss[r] = logf(red[0]) + m2 - lb[(size_t)t * QN];
}

// ---------------------------------------------------------------------------
// Kernel 3: deterministic final mean (single block, f64 accumulation)
// ---------------------------------------------------------------------------
__global__ void __launch_bounds__(TPB)
reduce_kernel(const float* __restrict__ rloss, const float* __restrict__ ploss,
              float* __restrict__ out)
{
  __shared__ double rs[TPB], ps[TPB];
  const int tid = threadIdx.x;
  double s1 = 0.0, s2 = 0.0;
  for (int i = tid; i < BS * NT; i += TPB) { s1 += rloss[i]; s2 += ploss[i]; }
  rs[tid] = s1; ps[tid] = s2; __syncthreads();
  for (int s = TPB / 2; s > 0; s >>= 1) {
    if (tid < s) { rs[tid] += rs[tid + s]; ps[tid] += ps[tid + s]; }
    __syncthreads();
  }
  if (tid == 0) {
    out[0] = (float)(rs[0] / (double)(BS * NT));
    out[1] = (float)(ps[0] / (double)(BS * NT));
  }
}

// ---------------------------------------------------------------------------
extern "C" void kernel_launch(void* const* d_in, const int* in_sizes, int n_in,
                              void* d_out, int out_size, void* d_ws, size_t ws_size,
                              hipStream_t stream) {
  (void)in_sizes; (void)n_in; (void)out_size; (void)ws_size;
  const float* rec     = (const float*)d_in[0];
  const float* loc     = (const float*)d_in[1];
  const int*   targets = (const int*)d_in[2];
  float*       out     = (float*)d_out;

  char* ws = (char*)d_ws;
  int*   match = (int*)ws;                  // BS*NT ints
  float* rloss = (float*)(ws + 16384);      // BS*NT floats
  float* ploss = (float*)(ws + 32768);      // BS*NT floats

  hungarian_kernel<<<dim3(BS), dim3(TPB), 0, stream>>>(rec, loc, targets, match);
  ce_kernel<<<dim3(BS * NT), dim3(CE_TPB), 0, stream>>>(rec, loc, targets, match,
                                                        rloss, ploss);
  reduce_kernel<<<dim3(1), dim3(TPB), 0, stream>>>(rloss, ploss, out);
}